// Rede2_56281251447333
// MI455X (gfx1250) — compile-verified
//
#include <hip/hip_runtime.h>
#include <math.h>

// ---------------------------------------------------------------------------
// MI455X (gfx1250, wave32) implementation.
//
// Roofline: conv1 = 5.8 GFLOP fp32 over 576 MB input (~10 FLOP/B). At
// 23.3 TB/s that needs > VALU fp32 peak, so conv1/conv2 use
// V_WMMA_F32_16X16X4_F32 (fp32-exact, matches reference precision) to get
// back to the HBM roof. Boundary handling is clamp-address + select (no
// exec-mask branches around loads -> clean global_load_b32 + v_cndmask).
// fc1 (64 x 912040 matvec, 233 MB weights) is pure bandwidth ->
// float4 streaming + deterministic two-phase block reduction.
// ---------------------------------------------------------------------------

typedef float v8f __attribute__((ext_vector_type(8)));
typedef float v2f __attribute__((ext_vector_type(2)));

#define XPLANE   (1200 * 1200)      // one input channel plane
#define POOL     300                 // 1200 / 4
#define PPLANE   (POOL * POOL)       // 90000
#define OX       302
#define PIX      (OX * OX)           // 91204
#define FLATN    (10 * PIX)          // 912040
#define FC1_CHUNK 8192
#define FC1_NCHUNK 112               // ceil(912040 / 8192)

static __device__ inline v8f wmma_f32x4(v2f a, v2f b, v8f c) {
  // D = A(16x4) * B(4x16) + C(16x16), fp32, wave32.
  return __builtin_amdgcn_wmma_f32_16x16x4_f32(
      false, a, false, b, (short)0, c, false, false);
}

// ---------------------------------------------------------------------------
// Kernel 1: conv1 (1x1, 10->20ch over zero-padded 1202^2) fused with 4x4 max
// pool. One wave produces pooled[bt][0..19][u][4vg..4vg+3].
//   A tiles: conv1_w (20x10) split M = {0..15, 16..19(+pad)}, K padded 10->12.
//   B tiles: 16 consecutive padded columns of one padded row.
//   Row-max is elementwise on D accumulators; column-max via shfl_xor(1,2).
// Grid is an exact wave count -> EXEC all-1s at every WMMA. Pad handling is
// clamped unconditional loads + selects (no divergent load branches).
// ---------------------------------------------------------------------------
__global__ void conv1_pool_kernel(const float* __restrict__ x,
                                  const float* __restrict__ w1,
                                  const float* __restrict__ b1,
                                  float* __restrict__ pooled) {
  const int lane  = threadIdx.x & 31;
  const int gwave = (blockIdx.x * blockDim.x + threadIdx.x) >> 5;
  // 10 frames * 300 pool rows * 75 groups-of-4 pool cols = 225000 waves exact
  const int bt  = gwave / (300 * 75);
  int rem       = gwave - bt * (300 * 75);
  const int u   = rem / 75;           // pool row
  const int vg  = rem - u * 75;       // group of 4 pool cols
  const int pc0 = vg * 16;            // first padded column
  const int half = lane >> 4;         // 0: lanes 0-15, 1: lanes 16-31
  const int n    = lane & 15;         // N index (pixel within tile)

  // A layout (16x4 fp32): v0 = K(2*half), v1 = K(2*half+1), row M = lane&15.
  v2f aLo[3], aHi[3];
  const int oLo = n;
  const int oHi = 16 + n;
  #pragma unroll
  for (int kt = 0; kt < 3; ++kt) {
    const int c0 = kt * 4 + half * 2;
    const int c1 = c0 + 1;
    aLo[kt].x = (c0 < 10) ? w1[oLo * 10 + c0] : 0.f;
    aLo[kt].y = (c1 < 10) ? w1[oLo * 10 + c1] : 0.f;
    aHi[kt].x = (oHi < 20 && c0 < 10) ? w1[oHi * 10 + c0] : 0.f;
    aHi[kt].y = (oHi < 20 && c1 < 10) ? w1[oHi * 10 + c1] : 0.f;
  }

  const float* xb = x + (long)bt * (10 * XPLANE);
  v8f mLo, mHi;
  #pragma unroll
  for (int g = 0; g < 8; ++g) { mLo[g] = -3.4e38f; mHi[g] = -3.4e38f; }

  const int  ocol  = pc0 + n - 1;             // original column (pad at -1)
  const int  colc  = (ocol >= 0) ? ocol : 0;  // clamped column
  const bool colok = (ocol >= 0);

  #pragma unroll
  for (int r = 0; r < 4; ++r) {
    const int  orow  = 4 * u + r - 1;         // original row (pad at -1)
    const bool pok   = (orow >= 0) && colok;
    const int  rowc  = (orow >= 0) ? orow : 0;
    const float* base = xb + rowc * 1200 + colc;

    // Gather all 6 B elements for this row unconditionally (clamped), then
    // zero the padded/overhang ones with selects.
    float bv[6];
    #pragma unroll
    for (int kt = 0; kt < 3; ++kt) {
      const int c0 = kt * 4 + half * 2;
      const int c1 = c0 + 1;
      const int c0c = (c0 < 10) ? c0 : 9;     // clamp channel (kt=2, half=1)
      const int c1c = (c1 < 10) ? c1 : 9;
      const float v0 = base[c0c * XPLANE];
      const float v1 = base[c1c * XPLANE];
      bv[2 * kt + 0] = (pok && c0 < 10) ? v0 : 0.f;
      bv[2 * kt + 1] = (pok && c1 < 10) ? v1 : 0.f;
    }

    v8f accLo = {0.f,0.f,0.f,0.f,0.f,0.f,0.f,0.f};
    v8f accHi = {0.f,0.f,0.f,0.f,0.f,0.f,0.f,0.f};
    #pragma unroll
    for (int kt = 0; kt < 3; ++kt) {
      v2f b;
      b.x = bv[2 * kt + 0];
      b.y = bv[2 * kt + 1];
      accLo = wmma_f32x4(aLo[kt], b, accLo);
      accHi = wmma_f32x4(aHi[kt], b, accHi);
    }
    #pragma unroll
    for (int g = 0; g < 8; ++g) {
      mLo[g] = fmaxf(mLo[g], accLo[g]);
      mHi[g] = fmaxf(mHi[g], accHi[g]);
    }
  }

  // Max over groups of 4 N-columns (one 4x4 pool block each).
  #pragma unroll
  for (int g = 0; g < 8; ++g) {
    float v = mLo[g];
    v = fmaxf(v, __shfl_xor(v, 1, 32));
    v = fmaxf(v, __shfl_xor(v, 2, 32));
    mLo[g] = v;
    float w = mHi[g];
    w = fmaxf(w, __shfl_xor(w, 1, 32));
    w = fmaxf(w, __shfl_xor(w, 2, 32));
    mHi[g] = w;
  }

  if ((n & 3) == 0) {
    const int v = vg * 4 + (n >> 2);  // pool column
    #pragma unroll
    for (int g = 0; g < 8; ++g) {
      const int m = g + 8 * half;     // channels 0..15
      pooled[((bt * 20 + m) * POOL + u) * POOL + v] = mLo[g] + b1[m];
      const int m2 = 16 + g + 8 * half; // channels 16..19 (rest padding)
      if (m2 < 20)
        pooled[((bt * 20 + m2) * POOL + u) * POOL + v] = mHi[g] + b1[m2];
    }
  }
}

// ---------------------------------------------------------------------------
// Kernel 2: zero-pad pooled (300->302) + conv2 (1x1, 20->10). One wave does
// one output row x 16 columns for all 10 outputs. K = 20 = 5 WMMA steps.
// Clamped unconditional loads + selects for the pad border.
// ---------------------------------------------------------------------------
__global__ void conv2_kernel(const float* __restrict__ pooled,
                             const float* __restrict__ w2,
                             const float* __restrict__ b2,
                             float* __restrict__ x5) {
  const int lane  = threadIdx.x & 31;
  const int gwave = (blockIdx.x * blockDim.x + threadIdx.x) >> 5;
  const int NW = 10 * OX * 19;        // 57380 waves
  if (gwave >= NW) return;            // wave-uniform guard

  const int bt  = gwave / (OX * 19);
  int rem       = gwave - bt * (OX * 19);
  const int ro  = rem / 19;           // output row 0..301
  const int ct  = rem - ro * 19;      // column tile
  const int col0 = ct * 16;
  const int half = lane >> 4;
  const int n    = lane & 15;
  const int o    = n;                 // output channel (valid < 10)

  v2f a[5];
  #pragma unroll
  for (int kt = 0; kt < 5; ++kt) {
    const int c0 = kt * 4 + half * 2;
    a[kt].x = (o < 10) ? w2[o * 20 + c0] : 0.f;
    a[kt].y = (o < 10) ? w2[o * 20 + c0 + 1] : 0.f;
  }

  const int col = col0 + n;
  const bool vb = (ro >= 1) && (ro <= 300) && (col >= 1) && (col <= 300);
  int rc = ro - 1;  if (rc < 0) rc = 0;  if (rc > 299) rc = 299;
  int cc = col - 1; if (cc < 0) cc = 0;  if (cc > 299) cc = 299;
  const float* base = pooled + (long)bt * (20 * PPLANE) + rc * POOL + cc;

  float bv[10];
  #pragma unroll
  for (int kt = 0; kt < 5; ++kt) {
    const int c0 = kt * 4 + half * 2;
    const float v0 = base[c0 * PPLANE];
    const float v1 = base[(c0 + 1) * PPLANE];
    bv[2 * kt + 0] = vb ? v0 : 0.f;
    bv[2 * kt + 1] = vb ? v1 : 0.f;
  }

  v8f acc = {0.f,0.f,0.f,0.f,0.f,0.f,0.f,0.f};
  #pragma unroll
  for (int kt = 0; kt < 5; ++kt) {
    v2f b;
    b.x = bv[2 * kt + 0];
    b.y = bv[2 * kt + 1];
    acc = wmma_f32x4(a[kt], b, acc);
  }

  if (col < OX) {
    #pragma unroll
    for (int g = 0; g < 8; ++g) {
      const int m = g + 8 * half;
      if (m < 10)
        x5[((bt * 10 + m) * OX + ro) * OX + col] = acc[g] + b2[m];
    }
  }
}

// ---------------------------------------------------------------------------
// Kernel 3: per-slot scalar LSTM + feature assembly.
// NOTE: the reference's dyn.reshape(P, T, ND) is a C-order re-fold of the
// (T, ND, OX, OY) block: xd[p, t, d] = dyn_flat[p*50 + t*5 + d], with
// dyn_flat[q] = x5[(t2*10 + d2)*PIX + pix] for q = t2*(5*PIX) + d2*PIX + pix.
// feat[c*PIX + p] = LSTM_h(p, d=c) for c<5; = x5[t=9, c] for c in 5..9.
// ---------------------------------------------------------------------------
__global__ void lstm_feat_kernel(const float* __restrict__ x5,
                                 const float* __restrict__ w_ih,
                                 const float* __restrict__ w_hh,
                                 const float* __restrict__ b_ih,
                                 const float* __restrict__ b_hh,
                                 float* __restrict__ feat) {
  const int tid = blockIdx.x * blockDim.x + threadIdx.x;
  if (tid >= 10 * PIX) return;
  const int c = tid / PIX;
  const int p = tid - c * PIX;

  if (c < 5) {
    const float wi0 = w_ih[c*4+0], wi1 = w_ih[c*4+1], wi2 = w_ih[c*4+2], wi3 = w_ih[c*4+3];
    const float wh0 = w_hh[c*4+0], wh1 = w_hh[c*4+1], wh2 = w_hh[c*4+2], wh3 = w_hh[c*4+3];
    const float bb0 = b_ih[c*4+0] + b_hh[c*4+0];
    const float bb1 = b_ih[c*4+1] + b_hh[c*4+1];
    const float bb2 = b_ih[c*4+2] + b_hh[c*4+2];
    const float bb3 = b_ih[c*4+3] + b_hh[c*4+3];
    float h = 0.f, cc = 0.f;
    for (int t = 0; t < 10; ++t) {
      const unsigned q = (unsigned)p * 50u + (unsigned)t * 5u + (unsigned)c;
      const unsigned t2  = q / (5u * PIX);
      const unsigned r   = q - t2 * (5u * PIX);
      const unsigned d2  = r / PIX;
      const unsigned pix = r - d2 * PIX;
      const float xv = x5[(t2 * 10u + d2) * PIX + pix];
      const float gi = xv * wi0 + h * wh0 + bb0;
      const float gf = xv * wi1 + h * wh1 + bb1;
      const float gg = xv * wi2 + h * wh2 + bb2;
      const float go = xv * wi3 + h * wh3 + bb3;
      const float ig = 1.f / (1.f + expf(-gi));
      const float fg = 1.f / (1.f + expf(-gf));
      const float g2 = tanhf(gg);
      const float og = 1.f / (1.f + expf(-go));
      cc = fg * cc + ig * g2;
      h  = og * tanhf(cc);
    }
    feat[c * PIX + p] = h;
  } else {
    feat[c * PIX + p] = x5[(90 + c) * PIX + p];   // stat = x5[t=9, ch 5..9]
  }
}

// ---------------------------------------------------------------------------
// Kernel 4a: fc1 partials. 112 blocks, each 8 waves; wave w owns outputs
// 8w..8w+7 over one K-chunk; lane-strided float4 loads (global_load_b128,
// coalesced weight rows), shuffle reduction. Fixed order -> deterministic.
// ---------------------------------------------------------------------------
__global__ void fc1_partial_kernel(const float* __restrict__ feat,
                                   const float* __restrict__ w,
                                   float* __restrict__ partials) {
  const int chunk = blockIdx.x;
  const int k0 = chunk * FC1_CHUNK;
  int kend = k0 + FC1_CHUNK;
  if (kend > FLATN) kend = FLATN;     // FLATN % 4 == 0, chunk base % 4 == 0
  const int wave = threadIdx.x >> 5;
  const int lane = threadIdx.x & 31;
  const int obase = wave * 8;

  float acc[8] = {0.f,0.f,0.f,0.f,0.f,0.f,0.f,0.f};
  for (int k = k0 + lane * 4; k < kend; k += 128) {
    const float4 xv = *(const float4*)(feat + k);
    #pragma unroll
    for (int j = 0; j < 8; ++j) {
      const float4 wv = *(const float4*)(w + (size_t)(obase + j) * FLATN + k);
      acc[j] += wv.x * xv.x + wv.y * xv.y + wv.z * xv.z + wv.w * xv.w;
    }
  }
  #pragma unroll
  for (int j = 0; j < 8; ++j) {
    float v = acc[j];
    for (int off = 16; off > 0; off >>= 1) v += __shfl_xor(v, off, 32);
    if (lane == 0) partials[chunk * 64 + obase + j] = v;
  }
}

// ---------------------------------------------------------------------------
// Kernel 4b: fold partials (fixed order) + fc2 -> relu -> fc3 -> fc4.
// ---------------------------------------------------------------------------
__global__ void fc_tail_kernel(const float* __restrict__ partials,
                               const float* __restrict__ fc1_b,
                               const float* __restrict__ fc2_w,
                               const float* __restrict__ fc2_b,
                               const float* __restrict__ fc3_w,
                               const float* __restrict__ fc3_b,
                               const float* __restrict__ fc4_w,
                               const float* __restrict__ fc4_b,
                               float* __restrict__ out, int nchunks) {
  __shared__ float y1[64], y2[64], y3[16];
  const int t = threadIdx.x;   // 64 threads
  float s = fc1_b[t];
  for (int cs = 0; cs < nchunks; ++cs) s += partials[cs * 64 + t];
  y1[t] = s;
  __syncthreads();
  float s2 = fc2_b[t];
  for (int j = 0; j < 64; ++j) s2 += fc2_w[t * 64 + j] * y1[j];
  y2[t] = fmaxf(s2, 0.f);
  __syncthreads();
  if (t < 16) {
    float s3 = fc3_b[t];
    for (int j = 0; j < 64; ++j) s3 += fc3_w[t * 64 + j] * y2[j];
    y3[t] = s3;
  }
  __syncthreads();
  if (t < 3) {
    float s4 = fc4_b[t];
    for (int j = 0; j < 16; ++j) s4 += fc4_w[t * 16 + j] * y3[j];
    out[t] = s4;
  }
}

// ---------------------------------------------------------------------------
extern "C" void kernel_launch(void* const* d_in, const int* in_sizes, int n_in,
                              void* d_out, int out_size, void* d_ws, size_t ws_size,
                              hipStream_t stream) {
  (void)in_sizes; (void)n_in; (void)out_size; (void)ws_size;
  const float* x    = (const float*)d_in[0];
  const float* w1   = (const float*)d_in[1];
  const float* b1   = (const float*)d_in[2];
  const float* w2   = (const float*)d_in[3];
  const float* b2   = (const float*)d_in[4];
  const float* wih  = (const float*)d_in[5];
  const float* whh  = (const float*)d_in[6];
  const float* bih  = (const float*)d_in[7];
  const float* bhh  = (const float*)d_in[8];
  const float* fc1w = (const float*)d_in[9];
  const float* fc1b = (const float*)d_in[10];
  const float* fc2w = (const float*)d_in[11];
  const float* fc2b = (const float*)d_in[12];
  const float* fc3w = (const float*)d_in[13];
  const float* fc3b = (const float*)d_in[14];
  const float* fc4w = (const float*)d_in[15];
  const float* fc4b = (const float*)d_in[16];
  float* out = (float*)d_out;

  // Workspace layout (floats):
  //   [0 .. 18,000,000)            pooled  (10*20*300*300)
  //   [18,000,000 .. 27,120,400)   x5      (10*10*302*302)
  //   feat reuses pooled[0 .. 912,040)      (after conv2 is done with pooled)
  //   partials at pooled + 1,000,000 (112*64)
  float* pooled   = (float*)d_ws;
  float* x5       = pooled + 18000000;
  float* feat     = pooled;
  float* partials = pooled + 1000000;

  // 225000 waves exact (no guard needed; full waves only)
  conv1_pool_kernel<<<28125, 256, 0, stream>>>(x, w1, b1, pooled);
  // 57380 waves -> 7173 blocks of 8 waves (wave-uniform guard inside)
  conv2_kernel<<<7173, 256, 0, stream>>>(pooled, w2, b2, x5);
  // 912040 threads
  lstm_feat_kernel<<<3563, 256, 0, stream>>>(x5, wih, whh, bih, bhh, feat);
  fc1_partial_kernel<<<FC1_NCHUNK, 256, 0, stream>>>(feat, fc1w, partials);
  fc_tail_kernel<<<1, 64, 0, stream>>>(partials, fc1b, fc2w, fc2b,
                                       fc3w, fc3b, fc4w, fc4b, out, FC1_NCHUNK);
}